// LatentFNO_24988119728193
// MI455X (gfx1250) — compile-verified
//
#include <hip/hip_runtime.h>
#include <math.h>

typedef __attribute__((ext_vector_type(2))) float v2f;
typedef __attribute__((ext_vector_type(8))) float v8f;

#define WIDTH   128
#define RR      512
#define MODES   64
#define BATCH   32
#define THIST   24
#define NLAYERS 4
#define NSTEPS  12
#define INCH    28

__device__ inline float gelu_exact(float x) {
  return 0.5f * x * (1.0f + erff(x * 0.70710678118654752f));
}

// One wave accumulates MT stacked 16x16 f32 tiles (a (16*MT) x 16 macro-tile):
//   acc[t] += A[16*t .. 16*t+15, :K] * B[:K, 0:16]
// The B fragment is loaded once per k-step and reused by all MT WMMAs.
// ISA layout: A: lanes 0-15 -> M=lane, K={0,1}; lanes 16-31 -> M=lane-16, K={2,3}.
//             B: lanes 0-15 -> K={0,1}, N=lane; lanes 16-31 -> K={2,3}, N=lane-16.
template <int MT>
__device__ inline void wmma_mt(v8f (&acc)[MT],
                               const float* __restrict__ A, int sAr, int sAc,
                               const float* __restrict__ B, int sBr, int sBc,
                               int K) {
  const int lane = threadIdx.x & 31;
  const int sel  = lane >> 4;
  const int rl   = lane & 15;
  const float* ap = A + rl * sAr + (2 * sel) * sAc;
  const float* bp = B + (2 * sel) * sBr + rl * sBc;
#pragma unroll 2
  for (int k = 0; k < K; k += 4) {
    v2f b;
    b.x = bp[0];
    b.y = bp[sBr];
#pragma unroll
    for (int t = 0; t < MT; ++t) {
      v2f a;
      a.x = ap[t * 16 * sAr];
      a.y = ap[t * 16 * sAr + sAc];
      acc[t] = __builtin_amdgcn_wmma_f32_16x16x4_f32(false, a, false, b,
                                                     (short)0, acc[t], false, false);
    }
    ap += 4 * sAc;
    bp += 4 * sBr;
  }
}

__global__ void k_copy(const float* __restrict__ src, float* __restrict__ dst, int n) {
  int i = blockIdx.x * blockDim.x + threadIdx.x;
  if (i < n) dst[i] = src[i];
}

// DFT basis Fc[r][0..63]=cos(2pi m r/512), Fc[r][64..127]=-sin(...)  (rfft)
// iDFT   G[m][r]=c_m cos/512, G[64+m][r]=-c_m sin/512  (irfft, modes 0..63)
__global__ void k_basis(float* __restrict__ Fc, float* __restrict__ G) {
  int i = blockIdx.x * blockDim.x + threadIdx.x;
  if (i >= RR * MODES) return;
  int r = i / MODES, m = i % MODES;
  int ph = (m * r) & (RR - 1);
  float ang = (float)ph * (6.283185307179586f / (float)RR);
  float cs = cosf(ang), sn = sinf(ang);
  Fc[r * 2 * MODES + m]         = cs;
  Fc[r * 2 * MODES + MODES + m] = -sn;
  float c = (m == 0) ? 1.0f : 2.0f;
  G[m * RR + r]           =  c * cs * (1.0f / (float)RR);
  G[(MODES + m) * RR + r] = -c * sn * (1.0f / (float)RR);
}

// IN[b,r,0:24]=seq history (ring), 24=sin24, 25=cos24, 26=sin168, 27=x_coord
__global__ void k_build_input(const float* __restrict__ seq,
                              const float* __restrict__ t_grid,
                              float* __restrict__ IN, int head, int step) {
  int i = blockIdx.x * blockDim.x + threadIdx.x;
  if (i >= BATCH * RR) return;
  int b = i >> 9, r = i & (RR - 1);
  float tt   = t_grid[b * THIST + (THIST - 1)] + (float)(step + 1);
  float a24  = 6.283185307179586f * tt * (1.0f / 24.0f);
  float a168 = 6.283185307179586f * tt * (1.0f / 168.0f);
  float* o = IN + (size_t)i * INCH;
#pragma unroll
  for (int t = 0; t < THIST; ++t) {
    int pr = head + t; if (pr >= THIST) pr -= THIST;
    o[t] = seq[(b * THIST + pr) * RR + r];
  }
  o[24] = sinf(a24);
  o[25] = cosf(a24);
  o[26] = sinf(a168);
  o[27] = (float)r * (1.0f / 511.0f);
}

// fc0: (B*R x 28) @ (28 x 128) + b  -> channel-major Hc[b, c, r]
__global__ __launch_bounds__(32) void k_fc0(const float* __restrict__ IN,
                                            const float* __restrict__ W,
                                            const float* __restrict__ bias,
                                            float* __restrict__ Hc) {
  int row0 = blockIdx.x * 16;       // rows = b*512 + r
  int n0   = blockIdx.y * 16;       // cols over 128
  v8f acc[1] = {};
  wmma_mt<1>(acc, IN + (size_t)row0 * INCH, INCH, 1, W + n0, WIDTH, 1, INCH);
  int lane = threadIdx.x & 31, sel = lane >> 4, rl = lane & 15;
  int n = n0 + rl;
  float bv = bias[n];
#pragma unroll
  for (int v = 0; v < 8; ++v) {
    int row = row0 + v + 8 * sel;
    int b = row >> 9, rr = row & (RR - 1);
    Hc[((size_t)(b * WIDTH + n)) * RR + rr] = acc[0][v] + bv;
  }
}

// forward DFT: X[(b*128+i), m2] = sum_r Hc[(b*128+i), r] * Fc[r, m2]
// Each wave computes a 64x16 macro-tile (4 WMMAs share one B fragment).
__global__ __launch_bounds__(32) void k_dft(const float* __restrict__ Hc,
                                            const float* __restrict__ Fc,
                                            float* __restrict__ X) {
  int row0 = blockIdx.x * 64;       // rows = b*128 + i (4096 -> 64 tiles)
  int n0   = blockIdx.y * 16;       // cols over 128 (= [Re 64 | Im 64])
  v8f acc[4] = {};
  wmma_mt<4>(acc, Hc + (size_t)row0 * RR, RR, 1, Fc + n0, 2 * MODES, 1, RR);
  int lane = threadIdx.x & 31, sel = lane >> 4, rl = lane & 15;
#pragma unroll
  for (int t = 0; t < 4; ++t)
#pragma unroll
    for (int v = 0; v < 8; ++v)
      X[((size_t)(row0 + 16 * t + v + 8 * sel)) * (2 * MODES) + n0 + rl] = acc[t][v];
}

// per-mode complex channel mix: Y = X @ W  (complex).
// One wave covers all 32 batch rows: the strided spectral-weight fragments
// (br, bi) are loaded once per k-step and feed 8 WMMAs.
__global__ __launch_bounds__(32) void k_mix(const float* __restrict__ X,
                                            const float* __restrict__ Wr,
                                            const float* __restrict__ Wi,
                                            float* __restrict__ Y) {
  int m  = blockIdx.x;              // mode
  int o0 = blockIdx.y * 16;         // out channels (128 -> 8 tiles)
  const int lane = threadIdx.x & 31, sel = lane >> 4, rl = lane & 15;
  // A (b,i):  X[(b*128+i)*128 + m]      -> b-stride 16384, i-stride 128
  // B (i,o):  W[(i*128+o)*64 + m]       -> i-stride 8192,  o-stride 64
  const float* ArB = X  + (size_t)rl * (WIDTH * 2 * MODES)
                        + (size_t)(2 * sel) * (2 * MODES) + m;
  const float* WrB = Wr + (size_t)(2 * sel) * (WIDTH * MODES)
                        + (size_t)(o0 + rl) * MODES + m;
  const float* WiB = Wi + (size_t)(2 * sel) * (WIDTH * MODES)
                        + (size_t)(o0 + rl) * MODES + m;
  const int bstep = 16 * (WIDTH * 2 * MODES);   // +16 batch rows
  v8f aRR[2] = {}, aII[2] = {}, aRI[2] = {}, aIR[2] = {};
#pragma unroll 2
  for (int k = 0; k < WIDTH; k += 4) {
    v2f br, bi;
    br.x = WrB[0];  br.y = WrB[WIDTH * MODES];
    bi.x = WiB[0];  bi.y = WiB[WIDTH * MODES];
#pragma unroll
    for (int t = 0; t < 2; ++t) {
      v2f ar, ai;
      const float* a = ArB + t * bstep;
      ar.x = a[0];      ar.y = a[2 * MODES];
      ai.x = a[MODES];  ai.y = a[2 * MODES + MODES];
      aRR[t] = __builtin_amdgcn_wmma_f32_16x16x4_f32(false, ar, false, br, (short)0, aRR[t], false, false);
      aII[t] = __builtin_amdgcn_wmma_f32_16x16x4_f32(false, ai, false, bi, (short)0, aII[t], false, false);
      aRI[t] = __builtin_amdgcn_wmma_f32_16x16x4_f32(false, ar, false, bi, (short)0, aRI[t], false, false);
      aIR[t] = __builtin_amdgcn_wmma_f32_16x16x4_f32(false, ai, false, br, (short)0, aIR[t], false, false);
    }
    ArB += 4 * (2 * MODES);
    WrB += 4 * (WIDTH * MODES);
    WiB += 4 * (WIDTH * MODES);
  }
#pragma unroll
  for (int t = 0; t < 2; ++t)
#pragma unroll
    for (int v = 0; v < 8; ++v) {
      int b = 16 * t + v + 8 * sel;
      int o = o0 + rl;
      size_t base = ((size_t)b * WIDTH + o) * (2 * MODES) + m;
      Y[base]         = aRR[t][v] - aII[t][v];   // real
      Y[base + MODES] = aRI[t][v] + aIR[t][v];   // imag
    }
}

// fused iDFT + pointwise conv + bias + exact GELU -> Hout[b, o, r]
// 64x16 macro-tile per wave, two contractions into the same accumulators.
__global__ __launch_bounds__(32) void k_spec_conv(const float* __restrict__ Y,
                                                  const float* __restrict__ G,
                                                  const float* __restrict__ Hin,
                                                  const float* __restrict__ CW,
                                                  const float* __restrict__ CB,
                                                  float* __restrict__ Hout) {
  int row0 = blockIdx.x * 64;       // rows = b*128 + o (4096 -> 64 tiles)
  int c0   = blockIdx.y * 16;       // cols over R=512
  int b = row0 >> 7, o0 = row0 & (WIDTH - 1);   // o0 in {0, 64}: same b
  v8f acc[4] = {};
  // spec: Y[(b*128+o), 0:128] @ G[0:128, r]
  wmma_mt<4>(acc, Y + (size_t)row0 * (2 * MODES), 2 * MODES, 1,
             G + c0, RR, 1, 2 * MODES);
  // pw:   CW[o, i] @ Hin[(b*128+i), r]
  wmma_mt<4>(acc, CW + (size_t)o0 * WIDTH, WIDTH, 1,
             Hin + (size_t)b * WIDTH * RR + c0, RR, 1, WIDTH);
  int lane = threadIdx.x & 31, sel = lane >> 4, rl = lane & 15;
#pragma unroll
  for (int t = 0; t < 4; ++t)
#pragma unroll
    for (int v = 0; v < 8; ++v) {
      int o = o0 + 16 * t + v + 8 * sel;
      float x = acc[t][v] + CB[o];
      Hout[((size_t)(b * WIDTH + o)) * RR + c0 + rl] = gelu_exact(x);
    }
}

// fc1: gather h back to (b,r,c) on the fly; GELU epilogue; 64x16 per wave
__global__ __launch_bounds__(32) void k_fc1(const float* __restrict__ Hc,
                                            const float* __restrict__ W,
                                            const float* __restrict__ bias,
                                            float* __restrict__ H2) {
  int row0 = blockIdx.x * 64;       // rows = b*512 + r (16384 -> 256 tiles)
  int n0   = blockIdx.y * 16;       // cols over 256
  int b = row0 >> 9, r0 = row0 & (RR - 1);      // 64 | 512: same b
  v8f acc[4] = {};
  // A (row=(b,r), k=c) = Hc[(b*128+k)*512 + r] -> row-stride 1, k-stride 512
  wmma_mt<4>(acc, Hc + (size_t)b * WIDTH * RR + r0, 1, RR,
             W + n0, 2 * WIDTH, 1, WIDTH);
  int lane = threadIdx.x & 31, sel = lane >> 4, rl = lane & 15;
  int n = n0 + rl;
  float bv = bias[n];
#pragma unroll
  for (int t = 0; t < 4; ++t)
#pragma unroll
    for (int v = 0; v < 8; ++v) {
      int row = row0 + 16 * t + v + 8 * sel;
      H2[(size_t)row * (2 * WIDTH) + n] = gelu_exact(acc[t][v] + bv);
    }
}

// fc2 (256 -> 1) + residual update, wave-per-row with coalesced loads and
// a wave32 shuffle reduction.
__global__ void k_fc2_update(const float* __restrict__ H2,
                             const float* __restrict__ W2,
                             const float* __restrict__ b2,
                             float* __restrict__ seq,
                             float* __restrict__ out, int head, int step) {
  int row  = blockIdx.x * (blockDim.x >> 5) + (threadIdx.x >> 5);
  int lane = threadIdx.x & 31;
  if (row >= BATCH * RR) return;
  const float* h = H2 + (size_t)row * (2 * WIDTH);
  float s = 0.0f;
#pragma unroll
  for (int n = lane; n < 2 * WIDTH; n += 32) s += h[n] * W2[n];
#pragma unroll
  for (int off = 16; off >= 1; off >>= 1) s += __shfl_xor(s, off, 32);
  if (lane == 0) {
    int b = row >> 9, r = row & (RR - 1);
    int last = head + THIST - 1; if (last >= THIST) last -= THIST;
    float zn = seq[(b * THIST + last) * RR + r] + s + b2[0];
    out[((size_t)b * NSTEPS + step) * RR + r] = zn;
    seq[(b * THIST + head) * RR + r] = zn;   // overwrite oldest ring slot
  }
}

extern "C" void kernel_launch(void* const* d_in, const int* in_sizes, int n_in,
                              void* d_out, int out_size, void* d_ws, size_t ws_size,
                              hipStream_t stream) {
  (void)in_sizes; (void)n_in; (void)out_size; (void)ws_size;
  const float* z     = (const float*)d_in[0];
  const float* tgrid = (const float*)d_in[1];
  const float* fc0_w = (const float*)d_in[2];
  const float* fc0_b = (const float*)d_in[3];
  const float* swr   = (const float*)d_in[4];
  const float* swi   = (const float*)d_in[5];
  const float* cw    = (const float*)d_in[6];
  const float* cb    = (const float*)d_in[7];
  const float* fc1_w = (const float*)d_in[8];
  const float* fc1_b = (const float*)d_in[9];
  const float* fc2_w = (const float*)d_in[10];
  const float* fc2_b = (const float*)d_in[11];
  float* out = (float*)d_out;

  float* ws  = (float*)d_ws;
  float* SEQ = ws; ws += BATCH * THIST * RR;       // ring buffer (B,24,512)
  float* Fc  = ws; ws += RR * 2 * MODES;           // forward DFT basis
  float* G   = ws; ws += 2 * MODES * RR;           // inverse DFT basis
  float* IN  = ws; ws += BATCH * RR * INCH;        // fc0 input (B,R,28)
  float* HcA = ws; ws += BATCH * WIDTH * RR;       // activations (B,128,512)
  float* HcB = ws; ws += BATCH * WIDTH * RR;
  float* X   = ws; ws += BATCH * WIDTH * 2 * MODES;// spectrum [Re|Im]
  float* Y   = ws; ws += BATCH * WIDTH * 2 * MODES;
  float* H2  = ws; ws += BATCH * RR * 2 * WIDTH;   // fc1 output (B,R,256)

  k_copy<<<(BATCH * THIST * RR + 255) / 256, 256, 0, stream>>>(z, SEQ, BATCH * THIST * RR);
  k_basis<<<(RR * MODES + 255) / 256, 256, 0, stream>>>(Fc, G);

  int head = 0;
  for (int s = 0; s < NSTEPS; ++s) {
    k_build_input<<<(BATCH * RR) / 256, 256, 0, stream>>>(SEQ, tgrid, IN, head, s);
    k_fc0<<<dim3((BATCH * RR) / 16, WIDTH / 16), 32, 0, stream>>>(IN, fc0_w, fc0_b, HcA);
    float* hin = HcA; float* hout = HcB;
    for (int l = 0; l < NLAYERS; ++l) {
      k_dft<<<dim3((BATCH * WIDTH) / 64, (2 * MODES) / 16), 32, 0, stream>>>(hin, Fc, X);
      k_mix<<<dim3(MODES, WIDTH / 16), 32, 0, stream>>>(
          X, swr + (size_t)l * WIDTH * WIDTH * MODES,
             swi + (size_t)l * WIDTH * WIDTH * MODES, Y);
      k_spec_conv<<<dim3((BATCH * WIDTH) / 64, RR / 16), 32, 0, stream>>>(
          Y, G, hin, cw + (size_t)l * WIDTH * WIDTH, cb + (size_t)l * WIDTH, hout);
      float* t = hin; hin = hout; hout = t;
    }
    k_fc1<<<dim3((BATCH * RR) / 64, (2 * WIDTH) / 16), 32, 0, stream>>>(hin, fc1_w, fc1_b, H2);
    k_fc2_update<<<(BATCH * RR) / 8, 256, 0, stream>>>(H2, fc2_w, fc2_b, SEQ, out, head, s);
    head = head + 1; if (head >= THIST) head = 0;
  }
}